// MedianBlur_79018808311964
// MI455X (gfx1250) — compile-verified
//
#include <hip/hip_runtime.h>

// 3x3 median blur, zero padding, fp32. B=16, C=3, H=W=512.
// Memory bound: ~96 MB total traffic -> ~4.1 us floor at 23.3 TB/s (MI455X).
// LDS-tiled (64x8 tile + 1px halo), Paeth 19-compare-exchange median network.
// Non-temporal stores keep the write stream from evicting re-read input lines.

#define TX 64
#define TY 8
#define LW (TX + 2)
#define LH (TY + 2)

__device__ __forceinline__ void ce(float& a, float& b) {
    float lo = fminf(a, b);
    float hi = fmaxf(a, b);
    a = lo;
    b = hi;
}

__global__ __launch_bounds__(TX * TY)
void MedianBlur_79018808311964_kernel(const float* __restrict__ in,
                                      float* __restrict__ out,
                                      int H, int W) {
    __shared__ float tile[LH][LW];

    const int bc = blockIdx.z;                      // fused batch*channel plane
    const long long base = (long long)bc * H * W;

    const int gx0 = blockIdx.x * TX - 1;            // tile origin incl. halo
    const int gy0 = blockIdx.y * TY - 1;

    const int tid = threadIdx.y * TX + threadIdx.x;

    // Cooperative, coalesced load of (TX+2)x(TY+2) tile; zero-fill halo OOB
    // to match jnp.pad zero padding. 660 elements / 512 threads = 2 iters.
    #pragma unroll
    for (int idx = tid; idx < LW * LH; idx += TX * TY) {
        const int ly = idx / LW;
        const int lx = idx - ly * LW;
        const int gx = gx0 + lx;
        const int gy = gy0 + ly;
        float v = 0.0f;
        if (gx >= 0 && gx < W && gy >= 0 && gy < H)
            v = in[base + (long long)gy * W + gx];
        tile[ly][lx] = v;
    }
    __syncthreads();

    const int tx = threadIdx.x;
    const int ty = threadIdx.y;

    float v0 = tile[ty + 0][tx + 0];
    float v1 = tile[ty + 0][tx + 1];
    float v2 = tile[ty + 0][tx + 2];
    float v3 = tile[ty + 1][tx + 0];
    float v4 = tile[ty + 1][tx + 1];
    float v5 = tile[ty + 1][tx + 2];
    float v6 = tile[ty + 2][tx + 0];
    float v7 = tile[ty + 2][tx + 1];
    float v8 = tile[ty + 2][tx + 2];

    // Paeth's 19-compare-exchange median-of-9 network; median lands in v4.
    ce(v1, v2); ce(v4, v5); ce(v7, v8);
    ce(v0, v1); ce(v3, v4); ce(v6, v7);
    ce(v1, v2); ce(v4, v5); ce(v7, v8);
    ce(v0, v3); ce(v5, v8); ce(v4, v7);
    ce(v3, v6); ce(v1, v4); ce(v2, v5);
    ce(v4, v7); ce(v4, v2); ce(v6, v4);
    ce(v4, v2);

    const int ox = blockIdx.x * TX + tx;
    const int oy = blockIdx.y * TY + ty;
    if (ox < W && oy < H)
        __builtin_nontemporal_store(v4, &out[base + (long long)oy * W + ox]);
}

extern "C" void kernel_launch(void* const* d_in, const int* in_sizes, int n_in,
                              void* d_out, int out_size, void* d_ws, size_t ws_size,
                              hipStream_t stream) {
    (void)n_in; (void)out_size; (void)d_ws; (void)ws_size;

    const float* image = (const float*)d_in[0];
    float* out = (float*)d_out;

    const int H = 512;
    const int W = 512;
    const int BC = in_sizes[0] / (H * W);   // 16 * 3 = 48 planes

    dim3 block(TX, TY, 1);
    dim3 grid((W + TX - 1) / TX, (H + TY - 1) / TY, BC);

    MedianBlur_79018808311964_kernel<<<grid, block, 0, stream>>>(image, out, H, W);
}